// EMADecomposition_8452495639009
// MI455X (gfx1250) — compile-verified
//
#include <hip/hip_runtime.h>
#include <hip/hip_bf16.h>

typedef float v2f __attribute__((ext_vector_type(2)));
typedef float v8f __attribute__((ext_vector_type(8)));

#define EMA_ALPHA 0.1f
#define BATCH 32
#define SEQ 4096
#define DMODEL 512
#define TILES (SEQ / 16)
#define SWAPX16 0x401F  // ds_swizzle group-of-32: xor=0x10, or=0, and=0x1f

// One wave owns one (batch, 16-wide d strip); scans 256 time-tiles of 16.
// Per tile: D(16x16) = T(16x16, lower-tri EMA weights) * X(16x16) + outer(w, carry)
// done as 4 chained V_WMMA_F32_16X16X4_F32.
__global__ void __launch_bounds__(256)
ema_decomp_wmma_kernel(const float* __restrict__ x,
                       float* __restrict__ trend,
                       float* __restrict__ seas) {
  const int lane  = threadIdx.x & 31;
  const int wave  = threadIdx.x >> 5;
  const int strip = blockIdx.x * 8 + wave;      // 1024 strips total
  const int b     = strip >> 5;                 // 512/16 = 32 strips per batch
  const int d0    = (strip & 31) << 4;
  const int n     = lane & 15;                  // column within strip
  const bool up   = (lane & 16) != 0;           // upper half-wave

  // ---- constant triangular EMA matrix T[i][c] = alpha*0.9^(i-c) (i>=c) in LDS.
  // Built once per block so per-lane A operands come from ds_load_b64 instead of
  // lane-indexed private arrays (which lower to 16-deep cndmask trees that the
  // compiler rematerializes inside the hot loop).
  __shared__ __align__(16) float Tlds[16 * 16];
  {
    const int idx = threadIdx.x;  // exactly 256 threads: one entry each
    const int i = idx >> 4;
    const int c = idx & 15;
    float v = 0.0f;
    if (i >= c) {
      v = EMA_ALPHA;
      for (int k = 0; k < i - c; ++k) v *= (1.0f - EMA_ALPHA);
    }
    Tlds[idx] = v;
  }
  __syncthreads();

  // powers of (1-alpha): only read at compile-time-constant indices -> folds.
  float pw[17];
  pw[0] = 1.0f;
#pragma unroll
  for (int k = 1; k <= 16; ++k) pw[k] = pw[k - 1] * (1.0f - EMA_ALPHA);

  // A-matrix slices (16x4 f32): lanes0-15 K={4j,4j+1}, lanes16-31 K={4j+2,4j+3}
  const int m  = n;                 // A row = lane&15
  const int kb = up ? 2 : 0;
  v2f A[4];
#pragma unroll
  for (int j = 0; j < 4; ++j) {
    const int c0 = 4 * j + kb;      // even -> 8B aligned LDS load
    A[j] = *reinterpret_cast<const v2f*>(&Tlds[m * 16 + c0]);
  }

  // carry weights per C/D row (row = r + (up?8:0)): w[row] = 0.9^(row+1)
  float wv[8];
#pragma unroll
  for (int r = 0; r < 8; ++r) wv[r] = up ? pw[r + 9] : pw[r + 1];

  // per-lane element base: (b, (up?8:0), d0+n)
  const size_t lanebase =
      (size_t)b * SEQ * DMODEL + (size_t)(up ? 8 : 0) * DMODEL + d0 + n;
  const float* px = x + lanebase;
  float* pt = trend + lanebase;
  float* ps = seas + lanebase;

  float cur[8], nxt[8];
#pragma unroll
  for (int r = 0; r < 8; ++r)
    cur[r] = __builtin_nontemporal_load(px + (size_t)r * DMODEL);

  float carry = 0.0f;  // initialized from x[0] at t==0

  for (int t = 0; t < TILES; ++t) {
    const size_t rowoff = (size_t)t * 16 * DMODEL;

    // software pipeline: next tile's 8 rows in flight over this tile's math
    if (t + 1 < TILES) {
      const size_t nrow = (size_t)(t + 1) * 16 * DMODEL;
#pragma unroll
      for (int r = 0; r < 8; ++r)
        nxt[r] = __builtin_nontemporal_load(px + nrow + (size_t)r * DMODEL);
    }

    // half-wave swaps: on lower lanes sw[r] = x[8+r][n]; on upper sw[r] = x[r][n]
    float sw[8];
#pragma unroll
    for (int r = 0; r < 8; ++r)
      sw[r] = __builtin_bit_cast(
          float, __builtin_amdgcn_ds_swizzle(__builtin_bit_cast(int, cur[r]),
                                             SWAPX16));

    if (t == 0) carry = up ? sw[0] : cur[0];  // trend[-1] := x[0]

    // C = outer(w, carry)
    v8f acc;
#pragma unroll
    for (int r = 0; r < 8; ++r) acc[r] = wv[r] * carry;

    // B operands, 4x16 f32 each: V0 lower=K(4j), upper=K(4j+2); V1 lower=K(4j+1), upper=K(4j+3)
    v2f B0, B1, B2, B3;
    B0.x = up ? sw[2]  : cur[0];  B0.y = up ? sw[3]  : cur[1];
    B1.x = up ? sw[6]  : cur[4];  B1.y = up ? sw[7]  : cur[5];
    B2.x = up ? cur[2] : sw[0];   B2.y = up ? cur[3] : sw[1];
    B3.x = up ? cur[6] : sw[4];   B3.y = up ? cur[7] : sw[5];

    // D = T*X + C  (K accumulated over 4 chained WMMAs)
    acc = __builtin_amdgcn_wmma_f32_16x16x4_f32(false, A[0], false, B0,
                                                (short)0, acc, false, false);
    acc = __builtin_amdgcn_wmma_f32_16x16x4_f32(false, A[1], false, B1,
                                                (short)0, acc, false, false);
    acc = __builtin_amdgcn_wmma_f32_16x16x4_f32(false, A[2], false, B2,
                                                (short)0, acc, false, false);
    acc = __builtin_amdgcn_wmma_f32_16x16x4_f32(false, A[3], false, B3,
                                                (short)0, acc, false, false);

    // streamed outputs: trend and seasonality = x - trend (cur IS x in D-layout)
#pragma unroll
    for (int r = 0; r < 8; ++r) {
      const float tr = acc[r];
      __builtin_nontemporal_store(tr, pt + rowoff + (size_t)r * DMODEL);
      __builtin_nontemporal_store(cur[r] - tr, ps + rowoff + (size_t)r * DMODEL);
    }

    // next carry = D row 15 (upper half of acc[7]), broadcast to both halves
    {
      const float a7 = acc[7];
      const float a7sw = __builtin_bit_cast(
          float,
          __builtin_amdgcn_ds_swizzle(__builtin_bit_cast(int, a7), SWAPX16));
      carry = up ? a7 : a7sw;
    }

#pragma unroll
    for (int r = 0; r < 8; ++r) cur[r] = nxt[r];
  }
}

extern "C" void kernel_launch(void* const* d_in, const int* in_sizes, int n_in,
                              void* d_out, int out_size, void* d_ws,
                              size_t ws_size, hipStream_t stream) {
  const float* x = (const float*)d_in[0];
  float* out = (float*)d_out;
  float* trend = out;                                   // first output
  float* seas = out + (size_t)BATCH * SEQ * DMODEL;     // second output

  // 1024 strips = 128 blocks * 8 waves, 256 threads/block (8 wave32)
  dim3 grid(128), block(256);
  hipLaunchKernelGGL(ema_decomp_wmma_kernel, grid, block, 0, stream, x, trend,
                     seas);
}